// SortMPNN_58171037057466
// MI455X (gfx1250) — compile-verified
//
#include <hip/hip_runtime.h>
#include <hip/hip_bf16.h>

typedef __attribute__((ext_vector_type(2))) float v2f;
typedef __attribute__((ext_vector_type(8))) float v8f;

// -----------------------------------------------------------------------------
// Kernel 1: xp[(n+1) x 64] = [x; blank_vec] @ W_proj^T + b_proj
// via V_WMMA_F32_16X16X4_F32 (exact fp32 semantics, 16 K-steps of 4).
// Block = 256 threads = 8 waves; wave (rt,ct) computes tile rows
// [m0+16*rt, +16) x cols [16*ct, +16). Grid covers n+1 rows.
// -----------------------------------------------------------------------------
__global__ __launch_bounds__(256) void sortmpnn_proj_wmma(
    const float* __restrict__ x,          // n x 64
    const float* __restrict__ blank_vec,  // 64
    const float* __restrict__ W,          // 64 x 64 (row-major, W_proj)
    const float* __restrict__ b_proj,     // 64
    float* __restrict__ xp,               // (n+1) x 64
    int n)
{
    const int lane  = threadIdx.x & 31;
    const int wave  = threadIdx.x >> 5;
    const int rt    = wave >> 2;           // 0..1
    const int ct    = wave & 3;            // 0..3
    const int m0    = blockIdx.x * 32 + rt * 16;
    const int n0    = ct * 16;
    const int lr    = lane & 15;           // M (for A) / N (for B,D) within tile
    const int khalf = (lane >> 4) * 2;     // lanes 16-31 hold K+2,K+3

    // A source row: real row, blank row (==n), or pad (>n -> read blank, never stored)
    const int arow = m0 + lr;
    const float* __restrict__ arp =
        (arow < n) ? (x + (size_t)arow * 64) : blank_vec;
    const float* __restrict__ brp = W + (size_t)(n0 + lr) * 64;

    v8f c = {0.f, 0.f, 0.f, 0.f, 0.f, 0.f, 0.f, 0.f};

#pragma unroll
    for (int k0 = 0; k0 < 64; k0 += 4) {
        const int ka = k0 + khalf;
        v2f a = *(const v2f*)(arp + ka);   // A[M=lr][ka], A[M=lr][ka+1]
        v2f b = *(const v2f*)(brp + ka);   // B[ka][N=n0+lr] = W[n0+lr][ka], +1
        c = __builtin_amdgcn_wmma_f32_16x16x4_f32(
                /*neg_a=*/false, a, /*neg_b=*/false, b,
                /*c_mod=*/(short)0, c, /*reuse_a=*/false, /*reuse_b=*/false);
    }

    // D layout: lanes 0-15 -> M=v, lanes 16-31 -> M=v+8; N = lane&15
    const int   col   = n0 + lr;
    const int   mbase = m0 + (lane >> 4) * 8;
    const float bias  = b_proj[col];
#pragma unroll
    for (int v = 0; v < 8; ++v) {
        const int m = mbase + v;
        if (m <= n) xp[(size_t)m * 64 + col] = c[v] + bias;
    }
}

// -----------------------------------------------------------------------------
// Kernel 2: one thread per (node, dim). Gather 16 messages from xp (L2-resident),
// sort in registers (bitonic-16), fold in the 16 implicit blank slots analytically,
// weighted-sum with w_collapse, and emit both halves of the concatenated output.
// Block = 256 threads = 4 nodes x 64 dims (coalesced gathers along dim).
// -----------------------------------------------------------------------------
__global__ __launch_bounds__(256) void sortmpnn_sort_collapse(
    const float* __restrict__ xp,          // (n+1) x 64 ; row n = blank_p
    const float* __restrict__ x,           // n x 64
    const float* __restrict__ blank_vec,   // 64
    const float* __restrict__ w_collapse,  // 32
    const float* __restrict__ b_collapse,  // 1
    const int*   __restrict__ src,         // n*16 edge sources
    float* __restrict__ out,               // (n+1) x 128
    int n)
{
    __shared__ float w_sh[32];
    __shared__ float wwin[17];   // wwin[k] = sum_{m=k}^{k+15} w_collapse[m]
    __shared__ int   src_sh[4 * 16];

    const int tid = threadIdx.x;
    if (tid < 32) w_sh[tid] = w_collapse[tid];
    if (tid < 17) {
        float s = 0.f;
#pragma unroll
        for (int m = 0; m < 16; ++m) s += w_collapse[tid + m];
        wwin[tid] = s;
    }
    const int node0 = blockIdx.x * 4;
    if (tid < 64) {
        const int ln = tid >> 4, e = tid & 15;
        const int nn = node0 + ln;
        // node n (blank row) and pad nodes: all slots point at blank row
        src_sh[tid] = (nn < n) ? src[nn * 16 + e] : n;
    }
    __syncthreads();

    const int ln = tid >> 6;     // 0..3
    const int d  = tid & 63;
    const int nn = node0 + ln;
    if (nn > n) return;

    const float blank = xp[(size_t)n * 64 + d];

    float v[16];
#pragma unroll
    for (int e = 0; e < 16; ++e)
        v[e] = xp[(size_t)src_sh[ln * 16 + e] * 64 + d];

    // Bitonic sort, ascending, 16 elements (80 compare-exchanges, all in VGPRs)
#pragma unroll
    for (int k = 2; k <= 16; k <<= 1) {
#pragma unroll
        for (int j = k >> 1; j > 0; j >>= 1) {
#pragma unroll
            for (int i = 0; i < 16; ++i) {
                const int l = i ^ j;
                if (l > i) {
                    const bool  up = ((i & k) == 0);
                    const float a = v[i], b = v[l];
                    const bool  sw = up ? (a > b) : (a < b);
                    v[i] = sw ? b : a;
                    v[l] = sw ? a : b;
                }
            }
        }
    }

    // k = #messages strictly below blank; blanks occupy merged slots [k, k+16)
    int kcnt = 0;
#pragma unroll
    for (int i = 0; i < 16; ++i) kcnt += (v[i] < blank) ? 1 : 0;

    float acc = b_collapse[0] + blank * wwin[kcnt];
#pragma unroll
    for (int i = 0; i < 16; ++i)
        acc += v[i] * ((v[i] < blank) ? w_sh[i] : w_sh[i + 16]);

    out[(size_t)nn * 128 + 64 + d] = acc;                               // aggr half
    out[(size_t)nn * 128 + d] = (nn < n) ? x[(size_t)nn * 64 + d]       // x half
                                         : blank_vec[d];
}

// -----------------------------------------------------------------------------
extern "C" void kernel_launch(void* const* d_in, const int* in_sizes, int n_in,
                              void* d_out, int out_size, void* d_ws, size_t ws_size,
                              hipStream_t stream) {
    const float* x          = (const float*)d_in[0];
    const float* blank_vec  = (const float*)d_in[1];
    const float* W_proj     = (const float*)d_in[2];
    const float* b_proj     = (const float*)d_in[3];
    const float* w_collapse = (const float*)d_in[4];
    const float* b_collapse = (const float*)d_in[5];
    const int*   edge_index = (const int*)d_in[6];   // (2, E): row0=dst, row1=src

    const int n = in_sizes[0] / 64;   // 50000
    const int E = in_sizes[6] / 2;    // n * 16
    const int* srcp = edge_index + E; // src row

    float* xp  = (float*)d_ws;        // (n+1) x 64 fp32 scratch (~12.8 MB)
    float* out = (float*)d_out;       // (n+1) x 128 fp32

    const int nb1 = (n + 1 + 31) / 32;
    sortmpnn_proj_wmma<<<nb1, 256, 0, stream>>>(x, blank_vec, W_proj, b_proj, xp, n);

    const int nb2 = (n + 1 + 3) / 4;
    sortmpnn_sort_collapse<<<nb2, 256, 0, stream>>>(xp, x, blank_vec, w_collapse,
                                                    b_collapse, srcp, out, n);
}